// KC_layer_73813307949286
// MI455X (gfx1250) — compile-verified
//
#include <hip/hip_runtime.h>
#include <hip/hip_bf16.h>

typedef __attribute__((ext_vector_type(2))) float v2f;
typedef __attribute__((ext_vector_type(8))) float v8f;

#define NFILT 8
#define SSUB  5
#define DIN   64
#define KSTEP 3
#define NCOLS 48                    // 8*5 = 40 cols padded to 3 tiles of 16
#define WHOP  (16*2*NCOLS*2)        // 3072 floats of B-fragment data per hop
#define INV_DTAO (1.0f/3.2f)        // 1/(DIN*TAO) = 1/(64*0.05)

#define WAVES 4
#define BPW   3                     // batches per wave (3*5=15 rows + pad row)
#define BPB   (WAVES*BPW)           // 12 batches per block

// ---------------------------------------------------------------------------
// Kernel 1: precompute hidden side once.
//  - A_h = symmetric sigmoid(adjs_hidden)
//  - fh[hop] = A_h^hop @ features_hidden
//  - Wg: WMMA-B-fragment-ready layout  (((hop*16+j)*2+kh)*48 + c)*2 + e
//        holding fh[hop][col c][k = 4j + 2*kh + e]  (c = 5*filter+s, 0 pad)
//  - sqHg[hop][c] = sum_k fh^2
// ---------------------------------------------------------------------------
__global__ void prep_hidden(const float* __restrict__ adjs_hidden,
                            const float* __restrict__ features_hidden,
                            float* __restrict__ Wg,
                            float* __restrict__ sqHg) {
  __shared__ float Ah[NFILT*SSUB*SSUB];
  __shared__ float fh[KSTEP][NFILT*SSUB*DIN];
  const int t = threadIdx.x;

  for (int i = t; i < NFILT*SSUB*SSUB; i += blockDim.x) {
    int f = i / 25, r = (i % 25) / 5, c = i % 5;
    float v = 0.0f;
    if (r != c) {
      int ii = r < c ? r : c, jj = r < c ? c : r;
      int p = ii*(9 - ii)/2 + (jj - ii - 1);       // upper-tri pair index
      float x = adjs_hidden[f*10 + p];
      v = 1.0f / (1.0f + __expf(-x));
    }
    Ah[i] = v;
  }
  for (int i = t; i < NFILT*SSUB*DIN; i += blockDim.x)
    fh[0][i] = features_hidden[i];
  __syncthreads();

  for (int h = 1; h < KSTEP; ++h) {
    for (int i = t; i < NFILT*SSUB*DIN; i += blockDim.x) {
      int f = i / (SSUB*DIN), r = (i / DIN) % SSUB, k = i % DIN;
      float acc = 0.0f;
#pragma unroll
      for (int j = 0; j < SSUB; ++j)
        acc += Ah[f*25 + r*5 + j] * fh[h-1][(f*SSUB + j)*DIN + k];
      fh[h][i] = acc;
    }
    __syncthreads();
  }

  for (int i = t; i < KSTEP*WHOP; i += blockDim.x) {
    int e  =  i & 1;
    int c  = (i >> 1) % NCOLS;
    int kh = (i / (NCOLS*2)) % 2;
    int j  = (i / (NCOLS*4)) % 16;
    int h  =  i / WHOP;
    float v = 0.0f;
    if (c < NFILT*SSUB) v = fh[h][c*DIN + (4*j + 2*kh + e)];
    Wg[i] = v;
  }
  for (int i = t; i < KSTEP*NCOLS; i += blockDim.x) {
    int h = i / NCOLS, c = i % NCOLS;
    float acc = 0.0f;
    if (c < NFILT*SSUB)
      for (int k = 0; k < DIN; ++k) { float x = fh[h][c*DIN + k]; acc += x*x; }
    sqHg[i] = acc;
  }
}

// ---------------------------------------------------------------------------
// Kernel 2: main. One wave handles 3 batches; cross terms via
// V_WMMA_F32_16X16X4_F32 over 3 column tiles of (filter,s) pairs.
// ---------------------------------------------------------------------------
__global__ __launch_bounds__(WAVES*32)
void kc_main(const float* __restrict__ adjs,
             const float* __restrict__ feature,
             const int*   __restrict__ idxs,
             const float* __restrict__ Wg,
             const float* __restrict__ sqHg,
             float* __restrict__ out, int Bn, int Nfeat) {
  __shared__ float Wl[WHOP];                 // B fragments, current hop
  __shared__ float sqHl[NCOLS];              // |fh|^2, current hop
  __shared__ float fb[WAVES][2][16*DIN];     // per-wave f ping-pong (16x64)
  __shared__ float sqFl[WAVES][16];          // per-wave |f row|^2

  const int tid  = threadIdx.x;
  const int w    = tid >> 5;
  const int lane = tid & 31;
  const int hi   = lane >> 4;                // K-half selector
  const int ln   = lane & 15;
  const int c0   = 2*lane;                   // this lane's pair of d-columns

  const int b_base = blockIdx.x * BPB + w * BPW;
  int  b_ld[BPW];
  bool valid[BPW];
#pragma unroll
  for (int g = 0; g < BPW; ++g) {
    int b = b_base + g;
    valid[g] = (b < Bn);
    b_ld[g]  = valid[g] ? b : (Bn - 1);      // clamp: never read OOB
  }

  // zero the pad row (M = 15) in both buffers
  fb[w][0][15*DIN + c0]     = 0.0f;
  fb[w][0][15*DIN + c0 + 1] = 0.0f;
  fb[w][1][15*DIN + c0]     = 0.0f;
  fb[w][1][15*DIN + c0 + 1] = 0.0f;

  // hop-0 gather: f[g*5+m][:] = feature[idxs[b][m]] (zero for pad / invalid)
#pragma unroll
  for (int g = 0; g < BPW; ++g) {
#pragma unroll
    for (int m = 0; m < SSUB; ++m) {
      int idx = idxs[b_ld[g]*SSUB + m];
      float x0 = 0.0f, x1 = 0.0f;
      if (valid[g] && idx >= 0 && idx < Nfeat) {
        const float* fp = feature + (size_t)idx*DIN + c0;
        x0 = fp[0]; x1 = fp[1];
      }
      fb[w][0][(g*SSUB + m)*DIN + c0]     = x0;
      fb[w][0][(g*SSUB + m)*DIN + c0 + 1] = x1;
    }
  }

  v8f tot[3] = {};

  for (int hop = 0; hop < KSTEP; ++hop) {
    // stage this hop's B fragments + squared norms (shared by all waves)
    for (int i = tid; i < WHOP; i += WAVES*32) Wl[i] = Wg[hop*WHOP + i];
    if (tid < NCOLS) sqHl[tid] = sqHg[hop*NCOLS + tid];

    const int cur = hop & 1;                 // 0,1,0
    if (hop > 0) {
      const int prev = cur ^ 1;
#pragma unroll
      for (int g = 0; g < BPW; ++g) {
        const float* ap = adjs + (size_t)b_ld[g]*(SSUB*SSUB);
        const float* fp = &fb[w][prev][g*SSUB*DIN + c0];
#pragma unroll
        for (int m = 0; m < SSUB; ++m) {
          float a0 = ap[m*5+0], a1 = ap[m*5+1], a2 = ap[m*5+2],
                a3 = ap[m*5+3], a4 = ap[m*5+4];
          float y0 = a0*fp[0*DIN]   + a1*fp[1*DIN]   + a2*fp[2*DIN]
                   + a3*fp[3*DIN]   + a4*fp[4*DIN];
          float y1 = a0*fp[0*DIN+1] + a1*fp[1*DIN+1] + a2*fp[2*DIN+1]
                   + a3*fp[3*DIN+1] + a4*fp[4*DIN+1];
          fb[w][cur][(g*SSUB + m)*DIN + c0]     = y0;
          fb[w][cur][(g*SSUB + m)*DIN + c0 + 1] = y1;
        }
      }
    }
    __syncthreads();                         // fb[cur] + Wl visible

    if (lane < 16) {                         // per-row |f|^2
      const float* fp = &fb[w][cur][lane*DIN];
      float acc = 0.0f;
      for (int k = 0; k < DIN; ++k) { float x = fp[k]; acc += x*x; }
      sqFl[w][lane] = acc;
    }
    __syncthreads();

    float sfv[8];
#pragma unroll
    for (int v = 0; v < 8; ++v) sfv[v] = sqFl[w][8*hi + v];

    const float* arow = &fb[w][cur][ln*DIN + 2*hi];   // A frag: K = 4j+2hi, +1
#pragma unroll
    for (int t3 = 0; t3 < 3; ++t3) {
      v8f acc = {};
      const float* brow = &Wl[(hi*NCOLS + (t3*16 + ln))*2];  // B frag base
#pragma unroll
      for (int j = 0; j < 16; ++j) {
        v2f av = *(const v2f*)(arow + 4*j);
        v2f bv = *(const v2f*)(brow + j*(NCOLS*4));          // stride 192
        acc = __builtin_amdgcn_wmma_f32_16x16x4_f32(
                  false, av, false, bv, (short)0, acc, false, false);
      }
      float sh = sqHl[t3*16 + ln];
#pragma unroll
      for (int v = 0; v < 8; ++v)
        tot[t3][v] += __expf((2.0f*acc[v] - sfv[v] - sh) * INV_DTAO);
    }
    __syncthreads();                         // done with Wl before restage
  }

  // dump total tiles to LDS as a dense 16x48 matrix per wave
  float* Cd = &fb[w][1][0];                  // reuse (768 <= 1024 floats)
#pragma unroll
  for (int t3 = 0; t3 < 3; ++t3)
#pragma unroll
    for (int v = 0; v < 8; ++v)
      Cd[(8*hi + v)*NCOLS + t3*16 + ln] = tot[t3][v];
  __syncthreads();

  // greedy argmax matching: one lane per (batch-in-wave, filter)
  if (lane < BPW*NFILT) {
    int g = lane / NFILT, F = lane % NFILT;
    if (valid[g]) {
      const float* T = &Cd[(g*SSUB)*NCOLS + F*SSUB];
      unsigned taken = 1u;                   // column 0 reserved for row 0
      float sum = T[0];
#pragma unroll
      for (int i = 1; i < SSUB; ++i) {
        const float* row = T + i*NCOLS;
        float best = -2.0f; int bj = 0;
#pragma unroll
        for (int jx = 0; jx < SSUB; ++jx) {
          float s = ((taken >> jx) & 1u) ? -1.0f : row[jx];
          if (s > best) { best = s; bj = jx; }   // strict > = first-max ties
        }
        taken |= (1u << bj);
        sum += row[bj];
      }
      out[(size_t)(b_base + g)*NFILT + F] = sum;
    }
  }
}

// ---------------------------------------------------------------------------
extern "C" void kernel_launch(void* const* d_in, const int* in_sizes, int n_in,
                              void* d_out, int out_size, void* d_ws, size_t ws_size,
                              hipStream_t stream) {
  const float* adjs            = (const float*)d_in[0];
  const float* feature         = (const float*)d_in[1];
  const int*   idxs            = (const int*)  d_in[2];
  const float* adjs_hidden     = (const float*)d_in[3];
  const float* features_hidden = (const float*)d_in[4];
  float* out = (float*)d_out;

  const int Bn    = in_sizes[0] / (SSUB*SSUB);
  const int Nfeat = in_sizes[1] / DIN;

  float* Wg   = (float*)d_ws;            // KSTEP*WHOP floats
  float* sqHg = Wg + KSTEP*WHOP;         // KSTEP*NCOLS floats (~37 KB total)

  prep_hidden<<<1, 256, 0, stream>>>(adjs_hidden, features_hidden, Wg, sqHg);

  const int nblk = (Bn + BPB - 1) / BPB;
  kc_main<<<nblk, WAVES*32, 0, stream>>>(adjs, feature, idxs, Wg, sqHg,
                                         out, Bn, Nfeat);
}